// GAT_RCA_50070728737104
// MI455X (gfx1250) — compile-verified
//
#include <hip/hip_runtime.h>
#include <hip/hip_bf16.h>

// ---------------------------------------------------------------------------
// GAT (2-layer, PyG-style) for gfx1250.
// GEMMs use V_WMMA_F32_16X16X4_F32 (exact fp32 matrix path on CDNA5).
// Edge phase: coalesced gather/scatter with atomic segment softmax.
// ---------------------------------------------------------------------------

typedef __attribute__((ext_vector_type(2))) float v2f;
typedef __attribute__((ext_vector_type(8))) float v8f;

#define NEG_SLOPE 0.2f
#define EPS 1e-16f

// ---------------- fp32 WMMA GEMM: C[M,Nout] = A[M,K] @ B[K,Nout] ------------
// Requirements: M % 16 == 0 (50000 = 3125*16), K % 4 == 0, Nout % 16 == 0.
// One wave computes one 16x16 tile of C, marching K in steps of 4.
// A 16x4 layout (ISA 7.12.2): lane l holds row M=l%16, regs {K=2*(l/16), +1}.
// B 4x16 layout: lane l holds col N=l%16, regs {K=2*(l/16), +1}.
// C/D: reg r, lane l -> C[r + 8*(l/16)][l%16].
__global__ void wmma_gemm_f32(const float* __restrict__ A,
                              const float* __restrict__ B,
                              float* __restrict__ C,
                              int M, int K, int Nout) {
    const int lane = threadIdx.x;          // 0..31
    const int lm   = lane & 15;
    const int lh   = lane >> 4;            // 0 or 1
    const int mt   = blockIdx.x;           // M tile
    const int nt   = threadIdx.y;          // N tile (blockDim.y = Nout/16)
    const int row0 = mt * 16;
    const int col0 = nt * 16;
    if (row0 >= M) return;

    const float* __restrict__ arow = A + (size_t)(row0 + lm) * K;
    v8f acc = {};
    for (int k0 = 0; k0 < K; k0 += 4) {
        const int ka = k0 + 2 * lh;
        v2f a, b;
        a.x = arow[ka];
        a.y = arow[ka + 1];
        b.x = B[(size_t)ka * Nout + col0 + lm];
        b.y = B[(size_t)(ka + 1) * Nout + col0 + lm];
        acc = __builtin_amdgcn_wmma_f32_16x16x4_f32(
            /*neg_a=*/false, a, /*neg_b=*/false, b,
            /*c_mod=*/(short)0, acc, /*reuse_a=*/false, /*reuse_b=*/false);
    }
    float* __restrict__ crow = C + (size_t)(row0 + 8 * lh) * Nout + col0 + lm;
#pragma unroll
    for (int r = 0; r < 8; ++r) crow[(size_t)r * Nout] = acc[r];
}

// ---------------- helpers ---------------------------------------------------
__device__ __forceinline__ void get_edge(const int* __restrict__ ei,
                                         long long idx, long long E,
                                         int& s, int& d) {
    if (idx < E) { s = ei[idx]; d = ei[E + idx]; }
    else         { s = d = (int)(idx - E); }        // self loop
}

// sign-aware float atomic max via integer atomics (lowers to native paths)
__device__ __forceinline__ void atomicMaxF(float* addr, float val) {
    if (val >= 0.0f) atomicMax((int*)addr, __float_as_int(val));
    else             atomicMin((unsigned int*)addr, __float_as_uint(val));
}

__global__ void fill_kernel(float* __restrict__ p, float v, long long n) {
    long long i = (long long)blockIdx.x * blockDim.x + threadIdx.x;
    if (i < n) p[i] = v;
}

// ---------------- per-node attention logits ---------------------------------
// alpha_s[n,h] = sum_c h[n,h,c]*a_src[h,c] ; alpha_d likewise
__global__ void alpha_kernel(const float* __restrict__ h,
                             const float* __restrict__ a_src,
                             const float* __restrict__ a_dst,
                             float* __restrict__ as, float* __restrict__ ad,
                             int n_nodes, int heads, int ch) {
    long long i = (long long)blockIdx.x * blockDim.x + threadIdx.x;
    if (i >= (long long)n_nodes * heads) return;
    const int head = (int)(i % heads);
    const long long node = i / heads;
    const float* __restrict__ hp  = h + ((size_t)node * heads + head) * ch;
    const float* __restrict__ asp = a_src + (size_t)head * ch;
    const float* __restrict__ adp = a_dst + (size_t)head * ch;
    float s = 0.f, d = 0.f;
    for (int c = 0; c < ch; ++c) { float v = hp[c]; s += v * asp[c]; d += v * adp[c]; }
    as[i] = s; ad[i] = d;
}

// ---------------- edge pass 1: logits + LeakyReLU + segment max -------------
__global__ void edge_logit_max(const int* __restrict__ ei, long long E, long long Et,
                               const float* __restrict__ as,
                               const float* __restrict__ ad,
                               float* __restrict__ e, float* __restrict__ m,
                               int heads) {
    long long i = (long long)blockIdx.x * blockDim.x + threadIdx.x;
    if (i >= Et * heads) return;
    const int head = (int)(i % heads);
    const long long eidx = i / heads;
    int s, d; get_edge(ei, eidx, E, s, d);
    float v = as[(size_t)s * heads + head] + ad[(size_t)d * heads + head];
    v = v > 0.f ? v : NEG_SLOPE * v;
    e[i] = v;
    atomicMaxF(&m[(size_t)d * heads + head], v);
}

// ---------------- edge pass 2: exp + segment denom --------------------------
__global__ void edge_exp_denom(const int* __restrict__ ei, long long E, long long Et,
                               float* __restrict__ e, const float* __restrict__ m,
                               float* __restrict__ den, int heads) {
    long long i = (long long)blockIdx.x * blockDim.x + threadIdx.x;
    if (i >= Et * heads) return;
    const int head = (int)(i % heads);
    const long long eidx = i / heads;
    int s, d; get_edge(ei, eidx, E, s, d);
    float ex = __expf(e[i] - m[(size_t)d * heads + head]);
    e[i] = ex;                                    // store un-normalized weight
    atomicAdd(&den[(size_t)d * heads + head], ex);
}

// ---------------- edge pass 3: weighted scatter aggregation -----------------
// thread per (edge, head, channel); 32 consecutive lanes = 1 contiguous row
__global__ void edge_agg(const int* __restrict__ ei, long long E, long long Et,
                         const float* __restrict__ e, const float* __restrict__ h,
                         float* __restrict__ agg, int heads, int ch) {
    long long i = (long long)blockIdx.x * blockDim.x + threadIdx.x;
    if (i >= Et * heads * ch) return;
    const int c    = (int)(i % ch);
    const int head = (int)((i / ch) % heads);
    const long long eidx = i / ((long long)ch * heads);
    int s, d; get_edge(ei, eidx, E, s, d);
    float v = e[eidx * heads + head] * h[((size_t)s * heads + head) * ch + c];
    atomicAdd(&agg[((size_t)d * heads + head) * ch + c], v);
}

// ---------------- node finalize, concat layout: (/denom) + bias + ELU -------
__global__ void finalize_concat(float* __restrict__ agg,
                                const float* __restrict__ den,
                                const float* __restrict__ bias,
                                int n_nodes, int heads, int ch) {
    long long i = (long long)blockIdx.x * blockDim.x + threadIdx.x;
    if (i >= (long long)n_nodes * heads * ch) return;
    const int c    = (int)(i % ch);
    const int head = (int)((i / ch) % heads);
    const long long node = i / ((long long)ch * heads);
    float v = agg[i] / (den[node * heads + head] + EPS) + bias[head * ch + c];
    agg[i] = v > 0.f ? v : (expf(v) - 1.0f);      // ELU, in place
}

// ---------------- node finalize, mean over heads + bias + ELU ---------------
__global__ void finalize_mean(const float* __restrict__ agg,
                              const float* __restrict__ den,
                              const float* __restrict__ bias,
                              float* __restrict__ out,
                              int n_nodes, int heads, int ch) {
    long long i = (long long)blockIdx.x * blockDim.x + threadIdx.x;
    if (i >= (long long)n_nodes * ch) return;
    const int c = (int)(i % ch);
    const long long node = i / ch;
    float acc = 0.f;
    for (int hh = 0; hh < heads; ++hh)
        acc += agg[((size_t)node * heads + hh) * ch + c] /
               (den[node * heads + hh] + EPS);
    float v = acc / (float)heads + bias[c];
    out[i] = v > 0.f ? v : (expf(v) - 1.0f);
}

// ---------------- final linear: out[n,o] = h[n,:] @ W[:,o] + b[o] -----------
__global__ void linear_out(const float* __restrict__ hf,
                           const float* __restrict__ W,
                           const float* __restrict__ b,
                           float* __restrict__ out,
                           int n_nodes, int inch, int outch) {
    long long i = (long long)blockIdx.x * blockDim.x + threadIdx.x;
    if (i >= (long long)n_nodes * outch) return;
    const int o = (int)(i % outch);
    const long long node = i / outch;
    const float* __restrict__ hp = hf + (size_t)node * inch;
    float acc = b[o];
    for (int c = 0; c < inch; ++c) acc += hp[c] * W[c * outch + o];
    out[i] = acc;
}

// ---------------------------------------------------------------------------
static inline unsigned gb(long long total, int bs) {
    return (unsigned)((total + bs - 1) / bs);
}

extern "C" void kernel_launch(void* const* d_in, const int* in_sizes, int n_in,
                              void* d_out, int out_size, void* d_ws, size_t ws_size,
                              hipStream_t stream) {
    const int IN_CH = 128, HID = 32, H1 = 4, H2 = 2, OUT = 3;
    const int N = in_sizes[0] / IN_CH;            // 50000
    const long long E  = in_sizes[1] / 2;         // 1.6M
    const long long Et = E + N;                   // + self loops

    const float* x     = (const float*)d_in[0];
    const int*   ei    = (const int*)  d_in[1];
    const float* W1    = (const float*)d_in[2];
    const float* asrc1 = (const float*)d_in[3];
    const float* adst1 = (const float*)d_in[4];
    const float* b1    = (const float*)d_in[5];
    const float* W2    = (const float*)d_in[6];
    const float* asrc2 = (const float*)d_in[7];
    const float* adst2 = (const float*)d_in[8];
    const float* b2    = (const float*)d_in[9];
    const float* Wlin  = (const float*)d_in[10];
    const float* blin  = (const float*)d_in[11];
    float* out = (float*)d_out;

    // workspace layout (floats)
    float* ws = (float*)d_ws;
    size_t off = 0;
    float* h1   = ws + off; off += (size_t)N * H1 * HID;   // [N,128]
    float* agg1 = ws + off; off += (size_t)N * H1 * HID;   // [N,128]; becomes h1a
    float* as_  = ws + off; off += (size_t)N * H1;         // reused by layer 2
    float* ad_  = ws + off; off += (size_t)N * H1;
    float* m_   = ws + off; off += (size_t)N * H1;
    float* den_ = ws + off; off += (size_t)N * H1;
    float* ebuf = ws + off; off += (size_t)Et * H1;        // reused by layer 2
    float* h2f  = ws + off; off += (size_t)N * HID;        // [N,32]
    // layer-2 buffers reuse the h1 region (N*128 >= N*64 + N*64)
    float* h2   = h1;                                      // [N,64]
    float* agg2 = h1 + (size_t)N * H2 * HID;               // [N,64]

    const int BS = 256;

    // ===== Layer 1 ==========================================================
    wmma_gemm_f32<<<dim3(N / 16, 1, 1), dim3(32, (H1 * HID) / 16, 1), 0, stream>>>(
        x, W1, h1, N, IN_CH, H1 * HID);
    alpha_kernel<<<gb((long long)N * H1, BS), BS, 0, stream>>>(
        h1, asrc1, adst1, as_, ad_, N, H1, HID);
    fill_kernel<<<gb((long long)N * H1, BS), BS, 0, stream>>>(m_, -INFINITY, (long long)N * H1);
    fill_kernel<<<gb((long long)N * H1, BS), BS, 0, stream>>>(den_, 0.f, (long long)N * H1);
    fill_kernel<<<gb((long long)N * H1 * HID, BS), BS, 0, stream>>>(agg1, 0.f, (long long)N * H1 * HID);
    edge_logit_max<<<gb(Et * H1, BS), BS, 0, stream>>>(ei, E, Et, as_, ad_, ebuf, m_, H1);
    edge_exp_denom<<<gb(Et * H1, BS), BS, 0, stream>>>(ei, E, Et, ebuf, m_, den_, H1);
    edge_agg<<<gb(Et * H1 * HID, BS), BS, 0, stream>>>(ei, E, Et, ebuf, h1, agg1, H1, HID);
    finalize_concat<<<gb((long long)N * H1 * HID, BS), BS, 0, stream>>>(
        agg1, den_, b1, N, H1, HID);                       // agg1 is now h (layer-2 input)

    // ===== Layer 2 ==========================================================
    wmma_gemm_f32<<<dim3(N / 16, 1, 1), dim3(32, (H2 * HID) / 16, 1), 0, stream>>>(
        agg1, W2, h2, N, H1 * HID, H2 * HID);
    alpha_kernel<<<gb((long long)N * H2, BS), BS, 0, stream>>>(
        h2, asrc2, adst2, as_, ad_, N, H2, HID);
    fill_kernel<<<gb((long long)N * H2, BS), BS, 0, stream>>>(m_, -INFINITY, (long long)N * H2);
    fill_kernel<<<gb((long long)N * H2, BS), BS, 0, stream>>>(den_, 0.f, (long long)N * H2);
    fill_kernel<<<gb((long long)N * H2 * HID, BS), BS, 0, stream>>>(agg2, 0.f, (long long)N * H2 * HID);
    edge_logit_max<<<gb(Et * H2, BS), BS, 0, stream>>>(ei, E, Et, as_, ad_, ebuf, m_, H2);
    edge_exp_denom<<<gb(Et * H2, BS), BS, 0, stream>>>(ei, E, Et, ebuf, m_, den_, H2);
    edge_agg<<<gb(Et * H2 * HID, BS), BS, 0, stream>>>(ei, E, Et, ebuf, h2, agg2, H2, HID);
    finalize_mean<<<gb((long long)N * HID, BS), BS, 0, stream>>>(
        agg2, den_, b2, h2f, N, H2, HID);

    // ===== Output linear ====================================================
    linear_out<<<gb((long long)N * OUT, BS), BS, 0, stream>>>(
        h2f, Wlin, blin, out, N, HID, OUT);
    (void)n_in; (void)out_size; (void)ws_size;
}